// BiDecoderRNN_32822140076387
// MI455X (gfx1250) — compile-verified
//
#include <hip/hip_runtime.h>
#include <math.h>

typedef __attribute__((ext_vector_type(2))) float v2f;
typedef __attribute__((ext_vector_type(8))) float v8f;

#define S_ 400
#define B_ 16
#define T_ 64
#define H_ 100
#define V_ 50000
#define A_ 25
#define VF_ 50
#define KP_ 52          // K padded to multiple of 4
#define KPAIRS_ (KP_/2) // 26
#define NROW (T_*B_)

// ---- output layout (floats) ----
#define OUT_PV   0
#define OUT_CL   (T_*B_*V_)
#define OUT_COV  (OUT_CL + T_*B_)
#define OUT_HID  (OUT_COV + S_*B_)
#define OUT_CELL (OUT_HID + 2*B_*H_)

// ---- workspace layout (floats) ----
#define WS_NEWENC  0
#define WS_NEWCELL (WS_NEWENC + 2*B_*H_)
#define WS_YS0     (WS_NEWCELL + 2*B_*H_)
#define WS_ODEC    (WS_YS0 + T_*B_*H_)
#define WS_ATTPRE  (WS_ODEC + T_*B_*H_)
#define WS_DTPROJ  (WS_ATTPRE + S_*B_*A_)
#define WS_COV     (WS_DTPROJ + T_*B_*A_)
#define WS_ATTN    (WS_COV + S_*B_)
#define WS_CTX     (WS_ATTN + T_*S_*B_)
#define WS_PG      (WS_CTX + T_*B_*2*H_)
#define WS_PVF     (WS_PG + NROW)            // padded [NROW][KP_]
#define WS_RMAX    (WS_PVF + NROW*KP_)
#define WS_RSUM    (WS_RMAX + NROW)
#define WS_WV2P    (WS_RSUM + NROW)          // packed [KPAIRS_][V_][2]

__device__ __forceinline__ float sigmoidf_(float x) { return 1.0f / (1.0f + __expf(-x)); }

// -------- zero --------
__global__ void k_zero(float* p, int n) {
  int i = blockIdx.x * blockDim.x + threadIdx.x;
  if (i < n) p[i] = 0.0f;
}

// -------- init: fold bidirectional encoder state: [2,B,2H] @ W_red.T + b --------
__global__ void k_init_reduce(const float* __restrict__ hid, const float* __restrict__ cell,
                              const float* __restrict__ Wr, const float* __restrict__ br,
                              float* __restrict__ ne, float* __restrict__ nc) {
  int idx = blockIdx.x * blockDim.x + threadIdx.x;
  if (idx >= 2 * 2 * B_ * H_) return;
  int which = idx / (2 * B_ * H_);
  int r = idx % (2 * B_ * H_);
  int l = r / (B_ * H_);
  int b = (r / H_) % B_;
  int h = r % H_;
  const float* src = which ? cell : hid;
  float acc = br[h];
  for (int k = 0; k < H_; ++k)
    acc += src[(l * B_ + b) * H_ + k] * Wr[h * 2 * H_ + k];
  for (int k = 0; k < H_; ++k)
    acc += src[((l + 2) * B_ + b) * H_ + k] * Wr[h * 2 * H_ + H_ + k];
  (which ? nc : ne)[(l * B_ + b) * H_ + h] = acc;
}

// -------- one LSTM layer, sequential over T, single workgroup, state in LDS --------
__global__ __launch_bounds__(1024)
void k_lstm(const float* __restrict__ x, int in_dim,
            const float* __restrict__ Wih, const float* __restrict__ Whh,
            const float* __restrict__ bih, const float* __restrict__ bhh,
            const float* __restrict__ h0, const float* __restrict__ c0,
            float* __restrict__ y, float* __restrict__ hT, float* __restrict__ cT) {
  __shared__ float sh[B_ * H_];
  __shared__ float sc[B_ * H_];
  __shared__ float sg[B_ * 4 * H_];
  const int tid = threadIdx.x, nt = blockDim.x;
  for (int i = tid; i < B_ * H_; i += nt) { sh[i] = h0[i]; sc[i] = c0[i]; }
  __syncthreads();
  for (int t = 0; t < T_; ++t) {
    for (int idx = tid; idx < B_ * 4 * H_; idx += nt) {
      int b = idx / (4 * H_), j = idx % (4 * H_);
      float acc = bih[j] + bhh[j];
      const float* xr = x + (t * B_ + b) * in_dim;
      for (int k = 0; k < in_dim; ++k) acc += xr[k] * Wih[j * in_dim + k];
      const float* hr = sh + b * H_;
      const float* wr = Whh + j * H_;
      for (int k = 0; k < H_; ++k) acc += hr[k] * wr[k];
      sg[idx] = acc;
    }
    __syncthreads();
    for (int idx = tid; idx < B_ * H_; idx += nt) {
      int b = idx / H_, jj = idx % H_;
      float gi = sg[b * 4 * H_ + jj];
      float gf = sg[b * 4 * H_ + H_ + jj];
      float gg = sg[b * 4 * H_ + 2 * H_ + jj];
      float go = sg[b * 4 * H_ + 3 * H_ + jj];
      float c_ = sigmoidf_(gf) * sc[idx] + sigmoidf_(gi) * tanhf(gg);
      float h_ = sigmoidf_(go) * tanhf(c_);
      sc[idx] = c_; sh[idx] = h_;
      y[t * B_ * H_ + idx] = h_;
    }
    __syncthreads();
  }
  for (int i = tid; i < B_ * H_; i += nt) { hT[i] = sh[i]; cT[i] = sc[i]; }
}

// -------- time-invariant attention precompute: Wa[:, :2H] @ output_enc + ba --------
__global__ void k_att_pre(const float* __restrict__ enc, const float* __restrict__ Wa,
                          const float* __restrict__ ba, float* __restrict__ pre) {
  int idx = blockIdx.x * blockDim.x + threadIdx.x;
  if (idx >= S_ * B_ * A_) return;
  int a = idx % A_;
  int sb = idx / A_;
  float acc = ba[a];
  const float* er = enc + sb * 2 * H_;
  const float* wr = Wa + a * (3 * H_ + 1);
  for (int k = 0; k < 2 * H_; ++k) acc += er[k] * wr[k];
  pre[idx] = acc;
}

// -------- per-step decoder hidden projection: Wa[:, 2H:3H] @ dt --------
__global__ void k_dt_proj(const float* __restrict__ odec, const float* __restrict__ Wa,
                          float* __restrict__ proj) {
  int idx = blockIdx.x * blockDim.x + threadIdx.x;
  if (idx >= T_ * B_ * A_) return;
  int a = idx % A_;
  int tb = idx / A_;
  float acc = 0.0f;
  const float* dr = odec + tb * H_;
  const float* wr = Wa + a * (3 * H_ + 1) + 2 * H_;
  for (int k = 0; k < H_; ++k) acc += dr[k] * wr[k];
  proj[idx] = acc;
}

// -------- one attention/coverage step; grid = B_ blocks --------
__global__ void k_att_step(int t, const float* __restrict__ pre, const float* __restrict__ proj,
                           const float* __restrict__ Wa, const float* __restrict__ Wo,
                           const float* __restrict__ amask, const float* __restrict__ enc,
                           float* __restrict__ cov, float* __restrict__ attn_all,
                           float* __restrict__ ctx, float* __restrict__ covloss) {
  __shared__ float se[S_];
  __shared__ float red[256];
  const int b = blockIdx.x, tid = threadIdx.x;
  for (int s = tid; s < S_; s += 256) {
    float cv = cov[s * B_ + b];
    float acc = 0.0f;
    for (int a = 0; a < A_; ++a) {
      float v = pre[(s * B_ + b) * A_ + a] + proj[(t * B_ + b) * A_ + a] +
                Wa[a * (3 * H_ + 1) + 3 * H_] * cv;
      acc += Wo[a] * tanhf(v);
    }
    se[s] = acc + amask[s * B_ + b];
  }
  __syncthreads();
  float m = -3.4e38f;
  for (int s = tid; s < S_; s += 256) m = fmaxf(m, se[s]);
  red[tid] = m; __syncthreads();
  for (int off = 128; off; off >>= 1) { if (tid < off) red[tid] = fmaxf(red[tid], red[tid + off]); __syncthreads(); }
  m = red[0]; __syncthreads();
  float z = 0.0f;
  for (int s = tid; s < S_; s += 256) z += __expf(se[s] - m);
  red[tid] = z; __syncthreads();
  for (int off = 128; off; off >>= 1) { if (tid < off) red[tid] += red[tid + off]; __syncthreads(); }
  float inv = 1.0f / red[0]; __syncthreads();
  float cl = 0.0f;
  for (int s = tid; s < S_; s += 256) {
    float a_ = __expf(se[s] - m) * inv;
    float cv = cov[s * B_ + b];
    cl += fminf(a_, cv);
    cov[s * B_ + b] = cv + a_;
    se[s] = a_;
    attn_all[(t * S_ + s) * B_ + b] = a_;
  }
  red[tid] = cl; __syncthreads();
  for (int off = 128; off; off >>= 1) { if (tid < off) red[tid] += red[tid + off]; __syncthreads(); }
  if (tid == 0) covloss[t * B_ + b] = red[0];
  __syncthreads();
  for (int j = tid; j < 2 * H_; j += 256) {
    float acc = 0.0f;
    for (int s = 0; s < S_; ++s) acc += se[s] * enc[(s * B_ + b) * 2 * H_ + j];
    ctx[(t * B_ + b) * 2 * H_ + j] = acc;
  }
}

// -------- pg = sigmoid([ctx, dt, it] @ Wp.T + bp) --------
__global__ void k_pg(const float* __restrict__ ctx, const float* __restrict__ odec,
                     const float* __restrict__ idec, const float* __restrict__ Wp,
                     const float* __restrict__ bp, float* __restrict__ pg) {
  int r = blockIdx.x * blockDim.x + threadIdx.x;
  if (r >= NROW) return;
  float acc = bp[0];
  for (int k = 0; k < 2 * H_; ++k) acc += ctx[r * 2 * H_ + k] * Wp[k];
  for (int k = 0; k < H_; ++k) acc += odec[r * H_ + k] * Wp[2 * H_ + k];
  acc += idec[r] * Wp[3 * H_];
  pg[r] = sigmoidf_(acc);
}

// -------- pv_feat = [dt, ctx] @ Wv1.T + bv1, written zero-padded to K=52 --------
__global__ void k_pvfeat(const float* __restrict__ odec, const float* __restrict__ ctx,
                         const float* __restrict__ Wv1, const float* __restrict__ bv1,
                         float* __restrict__ pvf) {
  int idx = blockIdx.x * blockDim.x + threadIdx.x;
  if (idx >= NROW * KP_) return;
  int r = idx / KP_, f = idx % KP_;
  if (f >= VF_) { pvf[idx] = 0.0f; return; }
  float acc = bv1[f];
  const float* wr = Wv1 + f * 3 * H_;
  for (int k = 0; k < H_; ++k) acc += odec[r * H_ + k] * wr[k];
  for (int k = 0; k < 2 * H_; ++k) acc += ctx[r * 2 * H_ + k] * wr[H_ + k];
  pvf[idx] = acc;
}

// -------- pack Wv2 [V,50] -> Wv2P [KPAIRS][V][2], zero-padded K -> coalesced b64 loads --------
__global__ void k_wv2pack(const float* __restrict__ Wv2, float* __restrict__ Wv2P) {
  int idx = blockIdx.x * blockDim.x + threadIdx.x;         // over KPAIRS_*V_
  if (idx >= KPAIRS_ * V_) return;
  int kp = idx / V_, v = idx % V_;
  int k0 = 2 * kp, k1 = 2 * kp + 1;
  float x = (k0 < VF_) ? Wv2[v * VF_ + k0] : 0.0f;
  float y = (k1 < VF_) ? Wv2[v * VF_ + k1] : 0.0f;
  v2f* dst = (v2f*)Wv2P;
  v2f p; p.x = x; p.y = y;
  dst[kp * V_ + v] = p;
}

// -------- batched vocab GEMM via f32 WMMA: [1024,52] x [52,50000] -> logits --------
// A tile 16x4 f32: lane m=lane&15; VGPR0=K(2*half), VGPR1=K(2*half)+1
// B tile 4x16 f32: lane n=lane&15; VGPR0=row K(2*half), VGPR1=row K(2*half)+1
// D 16x16: vgpr i -> row i + 8*half, col lane&15
__global__ __launch_bounds__(256)
void k_gemm_wmma(const float* __restrict__ pvf, const float* __restrict__ Wv2P,
                 const float* __restrict__ bv2, float* __restrict__ out) {
  const int lane = threadIdx.x & 31;
  const int wave = threadIdx.x >> 5;
  const int m_tile = blockIdx.y;                    // 0..63 (== t, since B_==16)
  const int n_tile = blockIdx.x * 8 + wave;         // 0..3124
  if (n_tile >= V_ / 16) return;                    // per-wave uniform; EXEC stays full
  const int half = lane >> 4;
  const int mrow = (m_tile << 4) + (lane & 15);
  const int n = (n_tile << 4) + (lane & 15);
  const v2f* __restrict__ Ap = (const v2f*)(pvf + mrow * KP_);
  const v2f* __restrict__ Bp = (const v2f*)Wv2P;
  v8f acc = {};
  #pragma unroll
  for (int kb = 0; kb < KP_; kb += 4) {
    const int kp = (kb >> 1) + half;                // K pair index for this lane half
    v2f a  = Ap[kp];                                // b64, pair (K, K+1)
    v2f bm = Bp[kp * V_ + n];                       // b64, coalesced across lanes
    acc = __builtin_amdgcn_wmma_f32_16x16x4_f32(false, a, false, bm,
                                                (short)0, acc, false, false);
  }
  const float bias = bv2[n];
  #pragma unroll
  for (int i = 0; i < 8; ++i) {
    int row = (m_tile << 4) + i + (half << 3);
    out[(long)row * V_ + n] = acc[i] + bias;
  }
}

// -------- per-row softmax stats over V (block per row) --------
__global__ void k_row_stats(const float* __restrict__ logits, float* __restrict__ rmax,
                            float* __restrict__ rsuminv) {
  __shared__ float red[256];
  const int r = blockIdx.x, tid = threadIdx.x;
  const float* row = logits + (long)r * V_;
  float m = -3.4e38f;
  for (int v = tid; v < V_; v += 256) m = fmaxf(m, row[v]);
  red[tid] = m; __syncthreads();
  for (int off = 128; off; off >>= 1) { if (tid < off) red[tid] = fmaxf(red[tid], red[tid + off]); __syncthreads(); }
  m = red[0]; __syncthreads();
  float z = 0.0f;
  for (int v = tid; v < V_; v += 256) z += __expf(row[v] - m);
  red[tid] = z; __syncthreads();
  for (int off = 128; off; off >>= 1) { if (tid < off) red[tid] += red[tid + off]; __syncthreads(); }
  if (tid == 0) { rmax[r] = m; rsuminv[r] = 1.0f / red[0]; }
}

// -------- out = exp(l - max)/sum * pg (in place; row = blockIdx.y, no int div) --------
__global__ void k_normalize(float* __restrict__ out, const float* __restrict__ rmax,
                            const float* __restrict__ rsuminv, const float* __restrict__ pg) {
  const int r = blockIdx.y;
  const int v = blockIdx.x * blockDim.x + threadIdx.x;
  if (v >= V_) return;
  const long i = (long)r * V_ + v;
  out[i] = __expf(out[i] - rmax[r]) * rsuminv[r] * pg[r];
}

// -------- copy distribution: scatter (1-pg)*attn at real_index --------
__global__ void k_scatter(const float* __restrict__ attn_all, const int* __restrict__ ridx,
                          const float* __restrict__ pg, float* __restrict__ out) {
  int idx = blockIdx.x * blockDim.x + threadIdx.x;
  if (idx >= T_ * S_ * B_) return;
  int t = idx / (S_ * B_);
  int rem = idx % (S_ * B_);
  int s = rem / B_, b = rem % B_;
  int v = ridx[s * B_ + b];
  int r = t * B_ + b;
  atomicAdd(&out[(long)r * V_ + v], (1.0f - pg[r]) * attn_all[idx]);
}

__global__ void k_copy(const float* __restrict__ src, float* __restrict__ dst, int n) {
  int i = blockIdx.x * blockDim.x + threadIdx.x;
  if (i < n) dst[i] = src[i];
}

extern "C" void kernel_launch(void* const* d_in, const int* in_sizes, int n_in,
                              void* d_out, int out_size, void* d_ws, size_t ws_size,
                              hipStream_t stream) {
  const float* output_enc = (const float*)d_in[0];
  const int*   real_index = (const int*)d_in[1];
  const float* input_dec  = (const float*)d_in[3];
  const float* hidden_enc = (const float*)d_in[4];
  const float* cell_enc   = (const float*)d_in[5];
  const float* att_mask   = (const float*)d_in[6];
  const float* W_red = (const float*)d_in[8];
  const float* b_red = (const float*)d_in[9];
  const float* Wih0 = (const float*)d_in[10];
  const float* Whh0 = (const float*)d_in[11];
  const float* bih0 = (const float*)d_in[12];
  const float* bhh0 = (const float*)d_in[13];
  const float* Wih1 = (const float*)d_in[14];
  const float* Whh1 = (const float*)d_in[15];
  const float* bih1 = (const float*)d_in[16];
  const float* bhh1 = (const float*)d_in[17];
  const float* Wa  = (const float*)d_in[18];
  const float* ba  = (const float*)d_in[19];
  const float* Wo  = (const float*)d_in[20];
  const float* Wv1 = (const float*)d_in[21];
  const float* bv1 = (const float*)d_in[22];
  const float* Wv2 = (const float*)d_in[23];
  const float* bv2 = (const float*)d_in[24];
  const float* Wp  = (const float*)d_in[25];
  const float* bp  = (const float*)d_in[26];

  float* out = (float*)d_out;
  float* ws  = (float*)d_ws;

  // coverage accumulator must start at zero every call
  k_zero<<<(S_ * B_ + 255) / 256, 256, 0, stream>>>(ws + WS_COV, S_ * B_);

  // one-time repack of Wv2 into padded, pair-major, transposed layout
  k_wv2pack<<<(KPAIRS_ * V_ + 255) / 256, 256, 0, stream>>>(Wv2, ws + WS_WV2P);

  // encoder-state reduction
  k_init_reduce<<<(2 * 2 * B_ * H_ + 255) / 256, 256, 0, stream>>>(
      hidden_enc, cell_enc, W_red, b_red, ws + WS_NEWENC, ws + WS_NEWCELL);

  // 2-layer LSTM (sequential in T inside the kernel)
  k_lstm<<<1, 1024, 0, stream>>>(input_dec, 1, Wih0, Whh0, bih0, bhh0,
                                 ws + WS_NEWENC, ws + WS_NEWCELL,
                                 ws + WS_YS0, out + OUT_HID, out + OUT_CELL);
  k_lstm<<<1, 1024, 0, stream>>>(ws + WS_YS0, H_, Wih1, Whh1, bih1, bhh1,
                                 ws + WS_NEWENC + B_ * H_, ws + WS_NEWCELL + B_ * H_,
                                 ws + WS_ODEC, out + OUT_HID + B_ * H_, out + OUT_CELL + B_ * H_);

  // hoisted attention projections
  k_att_pre<<<(S_ * B_ * A_ + 255) / 256, 256, 0, stream>>>(output_enc, Wa, ba, ws + WS_ATTPRE);
  k_dt_proj<<<(T_ * B_ * A_ + 255) / 256, 256, 0, stream>>>(ws + WS_ODEC, Wa, ws + WS_DTPROJ);

  // sequential coverage-attention loop (cheap per step)
  for (int t = 0; t < T_; ++t) {
    k_att_step<<<B_, 256, 0, stream>>>(t, ws + WS_ATTPRE, ws + WS_DTPROJ, Wa, Wo, att_mask,
                                       output_enc, ws + WS_COV, ws + WS_ATTN, ws + WS_CTX,
                                       out + OUT_CL);
  }

  // pg and padded vocab features for all steps at once
  k_pg<<<(NROW + 255) / 256, 256, 0, stream>>>(ws + WS_CTX, ws + WS_ODEC, input_dec, Wp, bp,
                                               ws + WS_PG);
  k_pvfeat<<<(NROW * KP_ + 255) / 256, 256, 0, stream>>>(ws + WS_ODEC, ws + WS_CTX, Wv1, bv1,
                                                         ws + WS_PVF);

  // batched vocab projection via f32 WMMA -> raw logits into pvocab output region
  dim3 ggrid((V_ / 16 + 7) / 8, T_);
  k_gemm_wmma<<<ggrid, 256, 0, stream>>>(ws + WS_PVF, ws + WS_WV2P, bv2, out + OUT_PV);

  // softmax over V, scale by pg, add copy distribution
  k_row_stats<<<NROW, 256, 0, stream>>>(out + OUT_PV, ws + WS_RMAX, ws + WS_RSUM);
  dim3 ngrid((V_ + 255) / 256, NROW);
  k_normalize<<<ngrid, 256, 0, stream>>>(out + OUT_PV, ws + WS_RMAX, ws + WS_RSUM, ws + WS_PG);
  k_scatter<<<(T_ * S_ * B_ + 255) / 256, 256, 0, stream>>>(ws + WS_ATTN, real_index, ws + WS_PG,
                                                            out + OUT_PV);

  // final coverage
  k_copy<<<(S_ * B_ + 255) / 256, 256, 0, stream>>>(ws + WS_COV, out + OUT_COV, S_ * B_);
}